// LightGCN_69329362092551
// MI455X (gfx1250) — compile-verified
//
#include <hip/hip_runtime.h>

#ifndef __has_builtin
#define __has_builtin(x) 0
#endif

#define N_USERS 100000
#define N_ITEMS 50000
#define N_NODES (N_USERS + N_ITEMS)
#define EMB 64
#define N_LAYERS 3
#define N_EDGES 2000000

#define CHUNK 1024        // edges staged to LDS per block
#define SCAT_BLOCK 256    // 8 waves; each wave owns one edge at a time

#if defined(__HIP_DEVICE_COMPILE__) && __has_builtin(__builtin_amdgcn_global_load_async_to_lds_b128)
#define USE_ASYNC_LDS 1
// Builtin expects pointers to a 16-byte int vector (per hipcc diagnostic);
// clang's CUDA/HIP builtin AS-rewrite converts generic -> AS1/AS3 for us.
typedef int lgcn_v4i __attribute__((vector_size(4 * sizeof(int))));
#else
#define USE_ASYNC_LDS 0
#endif

__device__ __forceinline__ void wait_asynccnt0() {
#if defined(__HIP_DEVICE_COMPILE__) && __has_builtin(__builtin_amdgcn_s_wait_asynccnt)
  __builtin_amdgcn_s_wait_asynccnt(0);
#elif defined(__HIP_DEVICE_COMPILE__)
  asm volatile("s_wait_asynccnt 0" ::: "memory");
#endif
}

__device__ __forceinline__ void atomic_add_f32(float* p, float v) {
  // relaxed, agent scope -> hardware global_atomic_add_f32 (no CAS loop)
  __hip_atomic_fetch_add(p, v, __ATOMIC_RELAXED, __HIP_MEMORY_SCOPE_AGENT);
}

// ---------------------------------------------------------------------------
// probe (defined FIRST, never launched): makes the f32 atomic lowering visible
// at the top of the asm snippet so we can confirm global_atomic_add_f32 vs CAS.
// ---------------------------------------------------------------------------
__global__ void lgcn_probe_atomic(float* p, float v) {
  atomic_add_f32(p + threadIdx.x, v);
}

// ---------------------------------------------------------------------------
// scatter: for each undirected edge (u, i):
//   next[i] += cur[u] * norm[u]      (u -> i message)
//   next[u] += cur[i] * norm[i]      (i -> u message)
// One block stages CHUNK edges' indices into LDS via async b128 copies,
// then each of the 8 waves processes edges; 32 lanes x float2 = one 64-wide row.
// ---------------------------------------------------------------------------
__global__ __launch_bounds__(SCAT_BLOCK) void lgcn_scatter(
    const float* __restrict__ cur, float* __restrict__ next,
    const float* __restrict__ norm,
    const int* __restrict__ eu, const int* __restrict__ ei) {
  __shared__ __align__(16) int s_u[CHUNK];
  __shared__ __align__(16) int s_i[CHUNK];

  const int chunkStart = (int)blockIdx.x * CHUNK;
  int rem = N_EDGES - chunkStart;
  const int cnt = rem < CHUNK ? rem : CHUNK;
  const int t = threadIdx.x;

  // --- stage edge indices into LDS (16B per lane per array) ---
  {
    int i4 = t * 4;
    if (i4 < cnt) {
      if (i4 + 4 <= cnt) {
#if USE_ASYNC_LDS
        __builtin_amdgcn_global_load_async_to_lds_b128(
            (lgcn_v4i*)(eu + chunkStart + i4), (lgcn_v4i*)(&s_u[i4]), 0, 0);
        __builtin_amdgcn_global_load_async_to_lds_b128(
            (lgcn_v4i*)(ei + chunkStart + i4), (lgcn_v4i*)(&s_i[i4]), 0, 0);
#else
        *(int4*)(&s_u[i4]) = *(const int4*)(eu + chunkStart + i4);
        *(int4*)(&s_i[i4]) = *(const int4*)(ei + chunkStart + i4);
#endif
      } else {
        for (int k = i4; k < cnt; ++k) {
          s_u[k] = eu[chunkStart + k];
          s_i[k] = ei[chunkStart + k];
        }
      }
    }
  }
#if USE_ASYNC_LDS
  wait_asynccnt0();
#endif
  __syncthreads();

  const int wave = t >> 5;   // 0..7
  const int lane = t & 31;
  const int col  = lane * 2; // float2 per lane covers EMB=64

  for (int j = wave; j < cnt; j += (SCAT_BLOCK / 32)) {
    // wave-uniform index reads from LDS -> scalar loads
    const int u  = s_u[j];
    const int it = s_i[j] + N_USERS;
    const float nu = norm[u];
    const float ni = norm[it];

    const float2 ru = *(const float2*)(cur + (size_t)u  * EMB + col);
    const float2 ri = *(const float2*)(cur + (size_t)it * EMB + col);

    float* du = next + (size_t)it * EMB + col;  // u -> it
    float* di = next + (size_t)u  * EMB + col;  // it -> u
    atomic_add_f32(du + 0, ru.x * nu);
    atomic_add_f32(du + 1, ru.y * nu);
    atomic_add_f32(di + 0, ri.x * ni);
    atomic_add_f32(di + 1, ri.y * ni);
  }
}

// ---------------------------------------------------------------------------
// init: bufA = h0 = concat(user_emb, item_emb); bufB = 0; out = h0; deg = 0
// ---------------------------------------------------------------------------
__global__ void lgcn_init(const float* __restrict__ ue, const float* __restrict__ ie,
                          float* __restrict__ bufA, float* __restrict__ bufB,
                          float* __restrict__ out, int* __restrict__ deg) {
  int i = blockIdx.x * blockDim.x + threadIdx.x;
  const int total = N_NODES * EMB;
  if (i < total) {
    float h = (i < N_USERS * EMB) ? ue[i] : ie[i - N_USERS * EMB];
    bufA[i] = h;
    bufB[i] = 0.0f;
    out[i]  = h;
  }
  if (i < N_NODES) deg[i] = 0;
}

// ---------------------------------------------------------------------------
// degree: one thread per undirected edge; both endpoints get +1
// ---------------------------------------------------------------------------
__global__ void lgcn_degree(const int* __restrict__ eu, const int* __restrict__ ei,
                            int* __restrict__ deg) {
  int e = blockIdx.x * blockDim.x + threadIdx.x;
  if (e < N_EDGES) {
    atomicAdd(&deg[eu[e]], 1);
    atomicAdd(&deg[N_USERS + ei[e]], 1);
  }
}

// ---------------------------------------------------------------------------
// norm[n] = deg>0 ? rsqrt(deg) : 0   (matches rsqrt(max(deg,1)) for deg>=1)
// ---------------------------------------------------------------------------
__global__ void lgcn_norm(const int* __restrict__ deg, float* __restrict__ norm) {
  int i = blockIdx.x * blockDim.x + threadIdx.x;
  if (i < N_NODES) {
    int d = deg[i];
    norm[i] = (d > 0) ? rsqrtf((float)d) : 0.0f;
  }
}

// ---------------------------------------------------------------------------
// node: h = norm[n] * h_io ; h_io = h (in place) ; out += h ; zero_buf = 0
// (zero_buf becomes the scatter target of the next layer)
// ---------------------------------------------------------------------------
__global__ void lgcn_node(float* __restrict__ h_io, float* __restrict__ zero_buf,
                          float* __restrict__ out, const float* __restrict__ norm) {
  int i = blockIdx.x * blockDim.x + threadIdx.x;
  if (i < N_NODES * EMB) {
    float h = h_io[i] * norm[i >> 6];
    h_io[i] = h;
    out[i] += h;
    zero_buf[i] = 0.0f;
  }
}

// out *= 1/(N_LAYERS+1)
__global__ void lgcn_final(float* __restrict__ out) {
  int i = blockIdx.x * blockDim.x + threadIdx.x;
  if (i < N_NODES * EMB) out[i] *= (1.0f / (float)(N_LAYERS + 1));
}

// ---------------------------------------------------------------------------
extern "C" void kernel_launch(void* const* d_in, const int* in_sizes, int n_in,
                              void* d_out, int out_size, void* d_ws, size_t ws_size,
                              hipStream_t stream) {
  const float* ue = (const float*)d_in[0];
  const float* ie = (const float*)d_in[1];
  const int*   eu = (const int*)d_in[2];
  const int*   ei = (const int*)d_in[3];
  float* out = (float*)d_out;

  // workspace layout
  char* ws = (char*)d_ws;
  const size_t HBYTES = (size_t)N_NODES * EMB * sizeof(float); // 38.4 MB
  int*   deg  = (int*)(ws);                         // N ints   @ 0
  float* norm = (float*)(ws + (1u << 20));          // N floats @ 1 MB
  float* bufA = (float*)(ws + (2u << 20));          // @ 2 MB
  float* bufB = (float*)(ws + (2u << 20) + HBYTES); // @ 2 MB + 38.4 MB

  const int total = N_NODES * EMB;
  const dim3 blk(256);

  lgcn_init<<<(total + 255) / 256, blk, 0, stream>>>(ue, ie, bufA, bufB, out, deg);
  lgcn_degree<<<(N_EDGES + 255) / 256, blk, 0, stream>>>(eu, ei, deg);
  lgcn_norm<<<(N_NODES + 255) / 256, blk, 0, stream>>>(deg, norm);

  float* cur = bufA;
  float* nxt = bufB;
  for (int l = 0; l < N_LAYERS; ++l) {
    lgcn_scatter<<<(N_EDGES + CHUNK - 1) / CHUNK, dim3(SCAT_BLOCK), 0, stream>>>(
        cur, nxt, norm, eu, ei);
    lgcn_node<<<(total + 255) / 256, blk, 0, stream>>>(nxt, cur, out, norm);
    float* tmp = cur; cur = nxt; nxt = tmp;
  }
  lgcn_final<<<(total + 255) / 256, blk, 0, stream>>>(out);
}